// SetAbstraction_10840497455586
// MI455X (gfx1250) — compile-verified
//
#include <hip/hip_runtime.h>
#include <cstdint>

// ---------------------------------------------------------------------------
// PointNet++ SetAbstraction for MI455X (gfx1250, wave32, WMMA).
// B=8, N=8192, S=2048, K=32, C_in=64, MLP=[64,64,128], radius^2=0.04.
// FLOP budget ~20G, HBM traffic ~1GB -> runtime dominated by the serial FPS
// chain + kNN selection, so GEMMs stay f32 via V_WMMA_F32_16X16X4_F32.
// ---------------------------------------------------------------------------

#define B_    8
#define N_    8192
#define CIN_  64
#define S_    2048
#define K_    32
#define M_    (B_*S_*K_)   // 524288 grouped points
#define R2_   0.04f
#define EPS_  1e-5f

typedef __attribute__((ext_vector_type(2))) float v2f;
typedef __attribute__((ext_vector_type(8))) float v8f;

__device__ __forceinline__ void argmax_combine(float& d, int& i, float od, int oi) {
  if (od > d || (od == d && oi < i)) { d = od; i = oi; }
}

// ---------------------------------------------------------------------------
// 1) Farthest point sampling: one workgroup per batch. Whole cloud (SoA) and
//    the running min-distance array live in LDS (CDNA5: 320KB/WGP).
//    2048 serial rounds of (distance update + block argmax).
// ---------------------------------------------------------------------------
__global__ __launch_bounds__(1024)
void fps_kernel(const float* __restrict__ xyz, int* __restrict__ fpsIdx,
                float* __restrict__ newXyz) {
  __shared__ float sx[N_], sy[N_], sz[N_], sd[N_];   // 128 KB
  __shared__ int   sidx[S_];                          // 8 KB
  __shared__ float rmax[32];
  __shared__ int   rarg[32];
  __shared__ int   sfar;
  const int b = blockIdx.x, tid = threadIdx.x;
  const int lane = tid & 31, wid = tid >> 5;

  for (int i = tid; i < N_; i += 1024) {
    const float* p = xyz + (size_t)(b*N_ + i)*3;
    sx[i] = p[0]; sy[i] = p[1]; sz[i] = p[2];
    sd[i] = 1e10f;
  }
  __syncthreads();

  int far = 0;
  for (int it = 0; it < S_; ++it) {
    if (tid == 0) sidx[it] = far;
    const float cx = sx[far], cy = sy[far], cz = sz[far];   // LDS broadcast
    float bd = -1.0f; int bi = 0;
    for (int i = tid; i < N_; i += 1024) {
      float dx = sx[i]-cx, dy = sy[i]-cy, dz = sz[i]-cz;
      float d  = dx*dx + dy*dy + dz*dz;
      float nd = fminf(sd[i], d);
      sd[i] = nd;
      if (nd > bd || (nd == bd && i < bi)) { bd = nd; bi = i; }
    }
    #pragma unroll
    for (int off = 16; off > 0; off >>= 1)
      argmax_combine(bd, bi, __shfl_down(bd, off), __shfl_down(bi, off));
    if (lane == 0) { rmax[wid] = bd; rarg[wid] = bi; }
    __syncthreads();
    if (wid == 0) {
      bd = rmax[lane]; bi = rarg[lane];
      #pragma unroll
      for (int off = 16; off > 0; off >>= 1)
        argmax_combine(bd, bi, __shfl_down(bd, off), __shfl_down(bi, off));
      if (lane == 0) sfar = bi;
    }
    __syncthreads();
    far = sfar;
  }

  for (int s = tid; s < S_; s += 1024) {
    const int j = sidx[s];
    fpsIdx[b*S_ + s] = j;
    float* o = newXyz + (size_t)(b*S_ + s)*3;
    o[0] = sx[j]; o[1] = sy[j]; o[2] = sz[j];
  }
}

// ---------------------------------------------------------------------------
// 2) Ball query / 32-NN: one wave32 per center. Each lane keeps a sorted
//    top-32 of its 256 strided points in a conflict-free LDS column
//    ([slot][lane] layout), then 32 shuffle-argmin merge rounds pick the
//    global 32 smallest (sorted ascending, matching lax.top_k of -dist).
//    Out-of-radius neighbors are replaced by the nearest one (idx[...,0]).
// ---------------------------------------------------------------------------
__global__ __launch_bounds__(256)
void ballquery_kernel(const float* __restrict__ xyz,
                      const float* __restrict__ newXyz,
                      int* __restrict__ grpIdx) {
  __shared__ float cd[8*32*32];   // 32 KB
  __shared__ int   ci[8*32*32];   // 32 KB
  const int lane = threadIdx.x & 31, w = threadIdx.x >> 5;
  const int gw = blockIdx.x*8 + w;
  const int b = gw / S_, s = gw % S_;
  const int base = w*1024 + lane;             // [w][slot][lane] = base + slot*32

  const float* ctr = newXyz + (size_t)(b*S_ + s)*3;
  const float cx = ctr[0], cy = ctr[1], cz = ctr[2];
  const float* xb = xyz + (size_t)b*N_*3;

  int cnt = 0;
  for (int p = 0; p < N_/32; ++p) {
    const int i = p*32 + lane;
    float dx = xb[i*3+0]-cx, dy = xb[i*3+1]-cy, dz = xb[i*3+2]-cz;
    float d  = dx*dx + dy*dy + dz*dz;
    if (cnt < 32 || d < cd[base + 31*32]) {
      int j = (cnt < 32) ? cnt : 31;
      while (j > 0 && cd[base + (j-1)*32] > d) {
        cd[base + j*32] = cd[base + (j-1)*32];
        ci[base + j*32] = ci[base + (j-1)*32];
        --j;
      }
      cd[base + j*32] = d;
      ci[base + j*32] = i;
      if (cnt < 32) ++cnt;
    }
  }

  int head = 0, firstIdx = 0;
  const size_t ob = (size_t)(b*S_ + s)*K_;
  for (int r = 0; r < K_; ++r) {
    float v  = (head < 32) ? cd[base + head*32] : 3.4e38f;
    int   id = (head < 32) ? ci[base + head*32] : 0x7fffffff;
    float bv = v; int bidx = id; int blane = lane;
    #pragma unroll
    for (int off = 16; off > 0; off >>= 1) {
      float ov = __shfl_down(bv, off);
      int   oi = __shfl_down(bidx, off);
      int   ol = __shfl_down(blane, off);
      if (ov < bv || (ov == bv && oi < bidx)) { bv = ov; bidx = oi; blane = ol; }
    }
    bv = __shfl(bv, 0); bidx = __shfl(bidx, 0); blane = __shfl(blane, 0);
    if (lane == blane) ++head;
    if (r == 0) firstIdx = bidx;
    if (lane == 0) grpIdx[ob + r] = (bv > R2_) ? firstIdx : bidx;
  }
}

// ---------------------------------------------------------------------------
// 3) Gather + concat into X0[68][M] (row 67 zero-padded so the WMMA K-loop
//    can step by 4 without guards). Coalesced writes along m.
// ---------------------------------------------------------------------------
__global__ __launch_bounds__(256)
void group_kernel(const float* __restrict__ xyz, const float* __restrict__ pts,
                  const float* __restrict__ newXyz, const int* __restrict__ grpIdx,
                  float* __restrict__ X0) {
  const int c = blockIdx.y;
  const int m = blockIdx.x*256 + threadIdx.x;
  const int b = m / (S_*K_);
  const int s = (m / K_) % S_;
  float val = 0.f;
  if (c < 67) {
    const int j = grpIdx[m];
    if (c < 3) val = xyz[(size_t)(b*N_ + j)*3 + c] - newXyz[(size_t)(b*S_ + s)*3 + c];
    else       val = pts[(size_t)(b*N_ + j)*CIN_ + (c - 3)];
  }
  X0[(size_t)c*M_ + m] = val;
}

// Pad weights [O][C] -> [O][Cp] (Cp multiple of 4, zero fill).
__global__ void padw_kernel(const float* __restrict__ w, float* __restrict__ wp,
                            int O, int C, int Cp) {
  const int t = blockIdx.x*256 + threadIdx.x;
  if (t < O*Cp) {
    const int o = t / Cp, c = t % Cp;
    wp[t] = (c < C) ? w[o*C + c] : 0.f;
  }
}

// ---------------------------------------------------------------------------
// 4) GEMM: Y[O][M] = W[O][Cp] * X[Cp][M] + bias, via V_WMMA_F32_16X16X4_F32.
//    Each wave: one 16(O) x 64(M) tile = 4 accumulators, A-fragment reused.
//    Fragment layouts per ISA 7.12.2:
//      A 16x4 : VGPR0 = {K0 | K2}, VGPR1 = {K1 | K3} across lane halves, row = lane&15
//      B 4x16 : VGPR0 = rows {0 | 2}, VGPR1 = rows {1 | 3}, col = lane&15
//      C 16x16: VGPR j = rows {j | j+8}, col = lane&15
// ---------------------------------------------------------------------------
__global__ __launch_bounds__(256)
void gemm_wmma_kernel(const float* __restrict__ W, const float* __restrict__ bias,
                      const float* __restrict__ X, float* __restrict__ Y,
                      int Otiles, int Cp) {
  const int wave = (blockIdx.x*256 + threadIdx.x) >> 5;
  const int lane = threadIdx.x & 31;
  const int mtiles = M_/64;
  const int ot = wave / mtiles;
  const int mt = wave % mtiles;
  if (ot >= Otiles) return;              // wave-uniform: EXEC stays all-ones
  const int o0 = ot*16, mb = mt*64;
  const int h = lane >> 4, n = lane & 15;

  v8f a0 = {}, a1 = {}, a2 = {}, a3 = {};
  for (int k = 0; k < Cp; k += 4) {
    v2f a;
    const float* wr = W + (o0 + n)*Cp + k + 2*h;
    a.x = wr[0]; a.y = wr[1];
    const float* x0 = X + (size_t)(k + 2*h)*M_ + mb + n;
    const float* x1 = x0 + M_;
    if (k + 4 < Cp) __builtin_prefetch(x0 + (size_t)4*M_, 0, 1);  // global_prefetch_b8
    v2f b0; b0.x = x0[0];  b0.y = x1[0];
    v2f b1; b1.x = x0[16]; b1.y = x1[16];
    v2f b2; b2.x = x0[32]; b2.y = x1[32];
    v2f b3; b3.x = x0[48]; b3.y = x1[48];
    a0 = __builtin_amdgcn_wmma_f32_16x16x4_f32(false, a, false, b0, (short)0, a0, false, false);
    a1 = __builtin_amdgcn_wmma_f32_16x16x4_f32(false, a, false, b1, (short)0, a1, false, false);
    a2 = __builtin_amdgcn_wmma_f32_16x16x4_f32(false, a, false, b2, (short)0, a2, false, false);
    a3 = __builtin_amdgcn_wmma_f32_16x16x4_f32(false, a, false, b3, (short)0, a3, false, false);
  }
  #pragma unroll
  for (int j = 0; j < 8; ++j) {
    const int row = o0 + j + 8*h;
    const float bb = bias[row];
    float* y = Y + (size_t)row*M_ + mb + n;
    y[0]  = a0[j] + bb;
    y[16] = a1[j] + bb;
    y[32] = a2[j] + bb;
    y[48] = a3[j] + bb;
  }
}

// ---------------------------------------------------------------------------
// 5) Training-mode BatchNorm: global per-channel sum / sumsq, then fused
//    normalize + ReLU in place.
// ---------------------------------------------------------------------------
__global__ void zero_stats_kernel(float* __restrict__ stats) {
  stats[threadIdx.x] = 0.f;
}

__global__ __launch_bounds__(256)
void stats_kernel(const float* __restrict__ X, float* __restrict__ stats) {
  const int c = blockIdx.x;
  const size_t base = (size_t)c*M_ + (size_t)blockIdx.y*8192;
  float s = 0.f, s2 = 0.f;
  for (int j = threadIdx.x; j < 8192; j += 256) {
    const float v = X[base + j];
    s += v; s2 += v*v;
  }
  #pragma unroll
  for (int off = 16; off > 0; off >>= 1) {
    s  += __shfl_down(s,  off);
    s2 += __shfl_down(s2, off);
  }
  __shared__ float r1[8], r2[8];
  if ((threadIdx.x & 31) == 0) { r1[threadIdx.x>>5] = s; r2[threadIdx.x>>5] = s2; }
  __syncthreads();
  if (threadIdx.x == 0) {
    float t1 = 0.f, t2 = 0.f;
    #pragma unroll
    for (int q = 0; q < 8; ++q) { t1 += r1[q]; t2 += r2[q]; }
    atomicAdd(&stats[2*c],     t1);
    atomicAdd(&stats[2*c + 1], t2);
  }
}

__global__ __launch_bounds__(256)
void bn_relu_kernel(float* __restrict__ X, const float* __restrict__ stats,
                    const float* __restrict__ g, const float* __restrict__ be) {
  const int c = blockIdx.y;
  const float invM = 1.0f / (float)M_;
  const float mean = stats[2*c] * invM;
  const float var  = stats[2*c + 1] * invM - mean*mean;   // biased, as jnp.var
  const float inv  = rsqrtf(var + EPS_);
  const float gg = g[c]*inv, bb = be[c] - mean*gg;
  const size_t i = (size_t)c*M_ + (size_t)blockIdx.x*256 + threadIdx.x;
  X[i] = fmaxf(X[i]*gg + bb, 0.f);
}

// ---------------------------------------------------------------------------
// 6) max over the S axis of [B,C,S,K] -> out[B,K,C]. One block per (b,c);
//    reads are fully coalesced (k = tid&31 is invariant under +=256 strides).
// ---------------------------------------------------------------------------
__global__ __launch_bounds__(256)
void maxpool_kernel(const float* __restrict__ X, float* __restrict__ out) {
  const int b = blockIdx.x >> 7;     // Cout = 128
  const int c = blockIdx.x & 127;
  const size_t base = (size_t)c*M_ + (size_t)b*S_*K_;
  float m = -3.4e38f;
  for (int j = threadIdx.x; j < S_*K_; j += 256) m = fmaxf(m, X[base + j]);
  __shared__ float red[256];
  red[threadIdx.x] = m;
  __syncthreads();
  if (threadIdx.x < 32) {
    float v = red[threadIdx.x];
    #pragma unroll
    for (int q = 1; q < 8; ++q) v = fmaxf(v, red[threadIdx.x + 32*q]);
    out[(size_t)(b*K_ + threadIdx.x)*128 + c] = v;
  }
}

// ---------------------------------------------------------------------------
extern "C" void kernel_launch(void* const* d_in, const int* in_sizes, int n_in,
                              void* d_out, int out_size, void* d_ws, size_t ws_size,
                              hipStream_t stream) {
  (void)in_sizes; (void)n_in; (void)out_size; (void)ws_size;
  const float* xyz = (const float*)d_in[0];
  const float* pts = (const float*)d_in[1];
  const float* W[3]  = {(const float*)d_in[2],  (const float*)d_in[6],  (const float*)d_in[10]};
  const float* Bi[3] = {(const float*)d_in[3],  (const float*)d_in[7],  (const float*)d_in[11]};
  const float* G[3]  = {(const float*)d_in[4],  (const float*)d_in[8],  (const float*)d_in[12]};
  const float* Be[3] = {(const float*)d_in[5],  (const float*)d_in[9],  (const float*)d_in[13]};

  float* newXyz = (float*)d_out;                        // [B,S,3]  = 49152 f32
  float* outPts = (float*)d_out + (size_t)B_*S_*3;      // [B,K,128]= 32768 f32

  // Workspace layout (~548 MB total).
  char* ws = (char*)d_ws;
  int*   fpsIdx = (int*)ws;     ws += (size_t)B_*S_*4;
  int*   grpIdx = (int*)ws;     ws += (size_t)M_*4;
  float* bufA   = (float*)ws;   ws += (size_t)68 *M_*4;  // X0 / layer1 out
  float* bufB   = (float*)ws;   ws += (size_t)64 *M_*4;  // layer0 out
  float* bufC   = (float*)ws;   ws += (size_t)128*M_*4;  // layer2 out
  float* wpad   = (float*)ws;   ws += (size_t)128*68*4;
  float* stats  = (float*)ws;

  fps_kernel      <<<B_,                 1024, 0, stream>>>(xyz, fpsIdx, newXyz);
  ballquery_kernel<<<(B_*S_)/8,           256, 0, stream>>>(xyz, newXyz, grpIdx);
  group_kernel    <<<dim3(M_/256, 68),    256, 0, stream>>>(xyz, pts, newXyz, grpIdx, bufA);

  const int Odim[3] = {64, 64, 128};
  const int Cdim[3] = {67, 64, 64};
  const int Cpad[3] = {68, 64, 64};
  float* Xin[3]  = {bufA, bufB, bufA};
  float* Yout[3] = {bufB, bufA, bufC};
  for (int l = 0; l < 3; ++l) {
    const int O = Odim[l], C = Cdim[l], P = Cpad[l];
    padw_kernel<<<(O*P + 255)/256, 256, 0, stream>>>(W[l], wpad, O, C, P);
    const int waves = (O/16) * (M_/64);
    gemm_wmma_kernel<<<waves/8, 256, 0, stream>>>(wpad, Bi[l], Xin[l], Yout[l], O/16, P);
    zero_stats_kernel<<<1, 256, 0, stream>>>(stats);
    stats_kernel<<<dim3(O, 64), 256, 0, stream>>>(Yout[l], stats);
    bn_relu_kernel<<<dim3(M_/256, O), 256, 0, stream>>>(Yout[l], stats, G[l], Be[l]);
  }

  maxpool_kernel<<<B_*128, 256, 0, stream>>>(bufC, outPts);
}